// PredictRetina2_38878043963704
// MI455X (gfx1250) — compile-verified
//
#include <hip/hip_runtime.h>
#include <stdint.h>

#define B_   32
#define A_   32526
#define C_   20
#define K_   1000
#define CONF_THR 0.5f
#define NMS_THR  0.5f
#define VAR0 0.1f
#define VAR1 0.2f
#define CLS_OFF 10.0f
#define KPAD 1024
#define TILES 63     // ceil(1000/16)
#define ROWW 64      // u16 halfwords per bit-matrix row (1024 cols)
#define P1_ANCH 256

typedef __attribute__((ext_vector_type(2))) float    v2f;
typedef __attribute__((ext_vector_type(8))) float    v8f;
typedef __attribute__((ext_vector_type(4))) uint32_t u32x4;
typedef __attribute__((ext_vector_type(8))) int      i32x8;
typedef __attribute__((ext_vector_type(4))) int      i32x4;

#if __has_builtin(__builtin_amdgcn_tensor_load_to_lds) && __has_builtin(__builtin_amdgcn_s_wait_tensorcnt)
#define USE_TDM 1
#endif
#if __has_builtin(__builtin_amdgcn_wmma_f32_16x16x4_f32)
#define USE_WMMA 1
#endif

__device__ __forceinline__ float fast_rcp(float x) {
#if __has_builtin(__builtin_amdgcn_rcpf)
  return __builtin_amdgcn_rcpf(x);   // v_rcp_f32, no div-fixup chain
#else
  return 1.0f / x;
#endif
}

// ---------------------------------------------------------------------------
// Phase 1: stream pcls tiles into LDS via Tensor Data Mover, reduce 20 classes
// to (sigmoid(max), argmax) per anchor.  sigmoid is monotone -> max of sigmoid
// == sigmoid of max logit; argmax of sigmoid == argmax of logits.
// ---------------------------------------------------------------------------
__global__ __launch_bounds__(P1_ANCH) void k_scores(
    const float* __restrict__ pcls, float* __restrict__ pscores, int* __restrict__ plabels)
{
  __shared__ float tile[P1_ANCH * C_];   // 20 KB
  const int b  = blockIdx.y;
  const int a0 = blockIdx.x * P1_ANCH;
  const int n  = min(P1_ANCH, A_ - a0);
  const float* gsrc = pcls + ((size_t)b * A_ + a0) * C_;
  const int nelem = n * C_;

#ifdef USE_TDM
  if (threadIdx.x < 32) {   // one wave issues the TDM descriptor (EXEC-independent op)
    uint32_t lds = (uint32_t)(uintptr_t)(&tile[0]);     // low 32 bits == LDS offset
    uint64_t ga  = (uint64_t)(uintptr_t)gsrc;
    u32x4 g0;
    g0[0] = 1u;                                         // count=1 valid descriptor
    g0[1] = lds;                                        // lds_addr
    g0[2] = (uint32_t)ga;                               // global_addr[31:0]
    g0[3] = (uint32_t)((ga >> 32) & 0x1FFFFFFu) | 0x80000000u; // ga[56:32] | type=2
    uint32_t td0 = (uint32_t)nelem;                     // 1-D tile: tensor_dim0 == tile_dim0
    i32x8 g1;
    g1[0] = (int)0x00020000u;                           // workgroup_mask=0, data_size=4B
    g1[1] = (int)((td0 & 0xFFFFu) << 16);               // tensor_dim0[15:0]
    g1[2] = (int)(((td0 >> 16) & 0xFFFFu) | (1u << 16));// tensor_dim0 hi | tensor_dim1=1
    g1[3] = (int)((td0 & 0xFFFFu) << 16);               // tile_dim0 (<=5120 fits 16b)
    g1[4] = (int)1u;                                    // tile_dim1=1, tile_dim2=0
    g1[5] = (int)td0;                                   // tensor_dim0_stride lo32
    g1[6] = 0; g1[7] = 0;
    i32x4 gz = {0, 0, 0, 0};
#if defined(__clang_major__) && (__clang_major__ >= 23)
    i32x8 z8 = {0, 0, 0, 0, 0, 0, 0, 0};
    __builtin_amdgcn_tensor_load_to_lds(g0, g1, gz, gz, z8, 0);
#else
    __builtin_amdgcn_tensor_load_to_lds(g0, g1, gz, gz, 0);
#endif
    __builtin_amdgcn_s_wait_tensorcnt(0);
  }
  __syncthreads();
#else
  for (int i = threadIdx.x; i < nelem; i += blockDim.x) tile[i] = gsrc[i];
  __syncthreads();
#endif

  const int t = threadIdx.x;
  if (t < n) {
    float v[C_];
#pragma unroll
    for (int c4 = 0; c4 < 5; ++c4) {               // 80B/thread stride -> 16B aligned
      float4 q = ((const float4*)(tile + t * C_))[c4];
      v[c4 * 4 + 0] = q.x; v[c4 * 4 + 1] = q.y;
      v[c4 * 4 + 2] = q.z; v[c4 * 4 + 3] = q.w;
    }
    float m = v[0]; int am = 0;
#pragma unroll
    for (int c = 1; c < C_; ++c) { if (v[c] > m) { m = v[c]; am = c; } }
    const float s = fast_rcp(1.0f + __expf(-m));   // sigmoid(max)
    const size_t o = (size_t)b * A_ + (a0 + t);
    pscores[o] = s;
    plabels[o] = am;
  }
}

// ---------------------------------------------------------------------------
// Phase 2: exact top-1000 per batch (radix-select over float bits, then
// bitonic sort of the 1000 winners, ties broken by smallest index like
// lax.top_k).  One 1024-thread block per batch.
// ---------------------------------------------------------------------------
__global__ __launch_bounds__(1024) void k_topk(
    const float* __restrict__ pscores, float* __restrict__ topv, int* __restrict__ topi)
{
  __shared__ unsigned hist[256];
  __shared__ unsigned s_prefix;
  __shared__ int      s_k;
  __shared__ unsigned s_cnt_gt, s_cnt_tie;
  __shared__ unsigned long long keys[KPAD];  // (scorebits<<32)|(~idx)
  __shared__ unsigned ties[2048];

  const int b = blockIdx.x;
  const int t = threadIdx.x;
  const float* sc = pscores + (size_t)b * A_;

  if (t == 0) { s_prefix = 0u; s_k = K_; }
  for (int pass = 0; pass < 4; ++pass) {          // sigmoid scores > 0 => bits monotone
    const int shift = 24 - 8 * pass;
    for (int i = t; i < 256; i += 1024) hist[i] = 0u;
    __syncthreads();
    const unsigned pref = s_prefix;
    for (int a = t; a < A_; a += 1024) {
      unsigned u = __float_as_uint(sc[a]);
      if ((unsigned)(((unsigned long long)u) >> (shift + 8)) == pref)
        atomicAdd(&hist[(u >> shift) & 255u], 1u);
    }
    __syncthreads();
    if (t == 0) {
      int k = s_k; unsigned cum = 0; int sel = 0;
      for (int d = 255; d >= 0; --d) {
        unsigned h = hist[d];
        if (cum + h >= (unsigned)k) { sel = d; break; }
        cum += h;
      }
      s_k = k - (int)cum;
      s_prefix = (pref << 8) | (unsigned)sel;
    }
    __syncthreads();
  }
  const unsigned T = s_prefix;
  int kt = s_k;
  if (t == 0) { s_cnt_gt = 0u; s_cnt_tie = 0u; }
  __syncthreads();
  for (int a = t; a < A_; a += 1024) {
    unsigned u = __float_as_uint(sc[a]);
    if (u > T) {
      unsigned pos = atomicAdd(&s_cnt_gt, 1u);
      if (pos < (unsigned)KPAD)
        keys[pos] = ((unsigned long long)u << 32) | (unsigned long long)(0xFFFFFFFFu - (unsigned)a);
    } else if (u == T) {
      unsigned pos = atomicAdd(&s_cnt_tie, 1u);
      if (pos < 2048u) ties[pos] = (unsigned)a;
    }
  }
  __syncthreads();
  unsigned cgt = s_cnt_gt; if (cgt > (unsigned)K_) cgt = (unsigned)K_;
  const unsigned ctie = (s_cnt_tie < 2048u) ? s_cnt_tie : 2048u;
  for (int i = t; i < 2048; i += 1024) if ((unsigned)i >= ctie) ties[i] = 0xFFFFFFFFu;
  __syncthreads();
  // bitonic sort ties ascending (smallest index first)
  for (int size = 2; size <= 2048; size <<= 1) {
    for (int stride = size >> 1; stride > 0; stride >>= 1) {
      for (int e = t; e < 2048; e += 1024) {
        int p = e ^ stride;
        if (p > e) {
          bool up = ((e & size) == 0);
          unsigned x = ties[e], y = ties[p];
          bool sw = up ? (x > y) : (x < y);
          if (sw) { ties[e] = y; ties[p] = x; }
        }
      }
      __syncthreads();
    }
  }
  if (kt < 0) kt = 0;
  {
    const unsigned need = (unsigned)K_ - cgt;
    if ((unsigned)kt > need) kt = (int)need;
  }
  for (int i = t; i < kt; i += 1024)
    keys[cgt + i] = ((unsigned long long)T << 32) | (unsigned long long)(0xFFFFFFFFu - ties[i]);
  for (int i = t; i < KPAD; i += 1024)
    if ((unsigned)i >= cgt + (unsigned)kt) keys[i] = 0ull;
  __syncthreads();
  // bitonic sort keys descending (score desc, index asc on ties)
  for (int size = 2; size <= KPAD; size <<= 1) {
    for (int stride = size >> 1; stride > 0; stride >>= 1) {
      int e = t, p = t ^ stride;
      if (p > e) {
        bool up = ((e & size) != 0);
        unsigned long long x = keys[e], y = keys[p];
        bool sw = up ? (x > y) : (x < y);
        if (sw) { keys[e] = y; keys[p] = x; }
      }
      __syncthreads();
    }
  }
  if (t < K_) {
    unsigned long long kk = keys[t];
    topv[(size_t)b * K_ + t] = __uint_as_float((unsigned)(kk >> 32));
    topi[(size_t)b * K_ + t] = (int)(0xFFFFFFFFu - (unsigned)(kk & 0xFFFFFFFFull));
  }
}

// ---------------------------------------------------------------------------
// Phase 3: decode selected boxes, labels, class-offset boxes, areas.
// ---------------------------------------------------------------------------
__global__ void k_decode(const float* __restrict__ ptxywh, const float* __restrict__ anchors,
                         const int* __restrict__ plabels, const int* __restrict__ topi,
                         float* __restrict__ boxes, float* __restrict__ oboxes,
                         float* __restrict__ areas, int* __restrict__ labels)
{
  int j = blockIdx.x * blockDim.x + threadIdx.x;
  if (j >= B_ * K_) return;
  const int b = j / K_;
  const int i = topi[j];
  const float* tp = ptxywh + ((size_t)b * A_ + i) * 4;
  const float* an = anchors + (size_t)i * 4;
  float ax = an[0], ay = an[1], aw = an[2], ah = an[3];
  float x = ax + tp[0] * VAR0 * aw;
  float y = ay + tp[1] * VAR0 * ah;
  float w = aw * __expf(tp[2] * VAR1);
  float h = ah * __expf(tp[3] * VAR1);
  float l = x - 0.5f * w, tt = y - 0.5f * h, r = x + 0.5f * w, bt = y + 0.5f * h;
  boxes[j * 4 + 0] = l;  boxes[j * 4 + 1] = tt;
  boxes[j * 4 + 2] = r;  boxes[j * 4 + 3] = bt;
  const int lab = plabels[(size_t)b * A_ + i] + 1;
  labels[j] = lab;
  const float off = (float)lab * CLS_OFF;
  oboxes[j * 4 + 0] = l + off;  oboxes[j * 4 + 1] = tt + off;
  oboxes[j * 4 + 2] = r + off;  oboxes[j * 4 + 3] = bt + off;
  areas[j] = (r - l) * (bt - tt);
}

// ---------------------------------------------------------------------------
// Phase 4a: 16x16 IoU tiles.  union-base tile (area_i + area_j) done as a
// rank-2 matmul with V_WMMA_F32_16X16X4_F32; intersection in VALU; division
// replaced by the exact equivalent inter > thr*max(union,eps); bits packed
// per row with wave32 ballot.  Each tile owns exclusive u16 row segments ->
// no atomics, no pre-zero needed.
// ---------------------------------------------------------------------------
__global__ __launch_bounds__(32) void k_pairs(
    const float* __restrict__ oboxes, const float* __restrict__ areas,
    uint16_t* __restrict__ mat)
{
  const int tj = blockIdx.x, ti = blockIdx.y, b = blockIdx.z;
  const int lane = threadIdx.x;
  const int half = lane >> 4, l16 = lane & 15;

  __shared__ float bi[16][4], bj[16][4], ai[16], aj[16];
  if (half == 0) {
    int ig = ti * 16 + l16;
    if (ig < K_) {
      const float* p = oboxes + ((size_t)b * K_ + ig) * 4;
      bi[l16][0] = p[0]; bi[l16][1] = p[1]; bi[l16][2] = p[2]; bi[l16][3] = p[3];
      ai[l16] = areas[(size_t)b * K_ + ig];
    } else {
      bi[l16][0] = bi[l16][1] = bi[l16][2] = bi[l16][3] = 0.f; ai[l16] = 0.f;
    }
  } else {
    int jg = tj * 16 + l16;
    if (jg < K_) {
      const float* p = oboxes + ((size_t)b * K_ + jg) * 4;
      bj[l16][0] = p[0]; bj[l16][1] = p[1]; bj[l16][2] = p[2]; bj[l16][3] = p[3];
      aj[l16] = areas[(size_t)b * K_ + jg];
    } else {
      bj[l16][0] = bj[l16][1] = bj[l16][2] = bj[l16][3] = 0.f; aj[l16] = 0.f;
    }
  }
  __syncthreads();

#ifdef USE_WMMA
  // A(16x4): col0 = area_i, col1 = 1;  B(4x16): row0 = 1, row1 = area_j.
  // D = A x B = area_i + area_j per 16x16 element (union base).
  v2f aV, bV;
  if (half == 0) { aV.x = ai[l16]; aV.y = 1.0f; bV.x = 1.0f; bV.y = aj[l16]; }
  else           { aV.x = 0.0f;    aV.y = 0.0f; bV.x = 0.0f; bV.y = 0.0f;   }
  v8f cz = {0.f, 0.f, 0.f, 0.f, 0.f, 0.f, 0.f, 0.f};
  v8f ub = __builtin_amdgcn_wmma_f32_16x16x4_f32(false, aV, false, bV, (short)0, cz,
                                                 false, false);
#endif

  const int N = l16, jg = tj * 16 + N;
  const float jl = bj[N][0], jt = bj[N][1], jr = bj[N][2], jb = bj[N][3];
  unsigned hw = 0;
#pragma unroll
  for (int v = 0; v < 8; ++v) {
    const int M = v + half * 8;
    const int ig = ti * 16 + M;
    float il = bi[M][0], it = bi[M][1], ir = bi[M][2], ib = bi[M][3];
    float wl = fmaxf(il, jl), wt = fmaxf(it, jt);
    float wr = fminf(ir, jr), wb = fminf(ib, jb);
    float inter = fmaxf(wr - wl, 0.f) * fmaxf(wb - wt, 0.f);
#ifdef USE_WMMA
    float uni = ub[v] - inter;
#else
    float uni = ai[M] + aj[N] - inter;
#endif
    // iou > thr  <=>  inter > thr * max(union, eps)   (division-free, exact)
    bool bit = (ig < K_) && (jg < K_) && (ig != jg) &&
               (inter > NMS_THR * fmaxf(uni, 1e-9f));
    unsigned m = (unsigned)__ballot(bit);   // wave32: [15:0]=row v, [31:16]=row v+8
    if (lane == v)     hw = m & 0xFFFFu;
    if (lane == v + 8) hw = (m >> 16) & 0xFFFFu;
  }
  if (lane < 16) {
    int ig = ti * 16 + lane;
    mat[((size_t)b * KPAD + ig) * ROWW + tj] = (uint16_t)hw;
  }
}

// ---------------------------------------------------------------------------
// Phase 4b: greedy NMS over the precomputed suppression bitmask.  One wave32
// per batch, each lane owns one 32-bit column word of the suppression vector.
// ---------------------------------------------------------------------------
__global__ __launch_bounds__(32) void k_greedy(const uint16_t* __restrict__ mat,
                                               const float* __restrict__ topv,
                                               uint8_t* __restrict__ keep)
{
  const int b = blockIdx.x, lane = threadIdx.x;
  __shared__ unsigned supp[32];
  __shared__ int kflag;
  __shared__ uint8_t ks[K_];
  supp[lane] = 0u;
  const uint32_t* rows = (const uint32_t*)(mat + (size_t)b * KPAD * ROWW);
  const float* tv = topv + (size_t)b * K_;
  const unsigned wmask = (lane == 31) ? 0x000000FFu : 0xFFFFFFFFu; // cols < 1000 only
  for (int i = 0; i < K_; ++i) {
    __syncthreads();
    if (lane == 0) {
      bool sup = (supp[i >> 5] >> (i & 31)) & 1u;
      kflag = (!sup && tv[i] >= CONF_THR) ? 1 : 0;
    }
    __syncthreads();
    if (kflag) supp[lane] |= rows[(size_t)i * 32 + lane] & wmask;
    if (lane == 0) ks[i] = (uint8_t)kflag;
  }
  __syncthreads();
  for (int i = lane; i < K_; i += 32) keep[(size_t)b * K_ + i] = ks[i];
}

// ---------------------------------------------------------------------------
// Phase 5: masked outputs, concatenated in reference return order.
// ---------------------------------------------------------------------------
__global__ void k_out(const float* __restrict__ topv, const int* __restrict__ labels,
                      const float* __restrict__ boxes, const uint8_t* __restrict__ keep,
                      float* __restrict__ out)
{
  int j = blockIdx.x * blockDim.x + threadIdx.x;
  if (j >= B_ * K_) return;
  const int b = j / K_;
  const float kf = keep[j] ? 1.0f : 0.0f;
  float* o_ids = out;
  float* o_box = o_ids + B_ * K_;
  float* o_lab = o_box + B_ * K_ * 4;
  float* o_sc  = o_lab + B_ * K_;
  float* o_kp  = o_sc + B_ * K_;
  o_ids[j] = (float)b;
  o_box[j * 4 + 0] = boxes[j * 4 + 0] * kf;
  o_box[j * 4 + 1] = boxes[j * 4 + 1] * kf;
  o_box[j * 4 + 2] = boxes[j * 4 + 2] * kf;
  o_box[j * 4 + 3] = boxes[j * 4 + 3] * kf;
  o_lab[j] = (float)labels[j];
  o_sc[j]  = topv[j] * kf;
  o_kp[j]  = kf;
}

extern "C" void kernel_launch(void* const* d_in, const int* in_sizes, int n_in,
                              void* d_out, int out_size, void* d_ws, size_t ws_size,
                              hipStream_t stream) {
  const float* ptxywh  = (const float*)d_in[0];
  const float* pcls    = (const float*)d_in[1];
  const float* anchors = (const float*)d_in[2];
  float* out = (float*)d_out;

  uint8_t* w = (uint8_t*)d_ws;
  auto take = [&](size_t bytes) -> uint8_t* {
    uint8_t* p = w; w += (bytes + 255) & ~(size_t)255; return p;
  };
  float*    pscores = (float*)   take((size_t)B_ * A_ * 4);
  int*      plabels = (int*)     take((size_t)B_ * A_ * 4);
  float*    topv    = (float*)   take((size_t)B_ * K_ * 4);
  int*      topi    = (int*)     take((size_t)B_ * K_ * 4);
  float*    boxes   = (float*)   take((size_t)B_ * K_ * 16);
  float*    oboxes  = (float*)   take((size_t)B_ * K_ * 16);
  float*    areas   = (float*)   take((size_t)B_ * K_ * 4);
  int*      labels  = (int*)     take((size_t)B_ * K_ * 4);
  uint8_t*  keep    = (uint8_t*) take((size_t)B_ * K_);
  uint16_t* mat     = (uint16_t*)take((size_t)B_ * KPAD * ROWW * 2);
  (void)ws_size; (void)in_sizes; (void)n_in; (void)out_size;

  dim3 g1((A_ + P1_ANCH - 1) / P1_ANCH, B_);
  k_scores<<<g1, P1_ANCH, 0, stream>>>(pcls, pscores, plabels);

  k_topk<<<B_, 1024, 0, stream>>>(pscores, topv, topi);

  const int n = B_ * K_;
  k_decode<<<(n + 255) / 256, 256, 0, stream>>>(ptxywh, anchors, plabels, topi,
                                                boxes, oboxes, areas, labels);

  dim3 gp(TILES, TILES, B_);
  k_pairs<<<gp, 32, 0, stream>>>(oboxes, areas, mat);

  k_greedy<<<B_, 32, 0, stream>>>(mat, topv, keep);

  k_out<<<(n + 255) / 256, 256, 0, stream>>>(topv, labels, boxes, keep, out);
}